// RandomSortCM_36386962931768
// MI455X (gfx1250) — compile-verified
//
#include <hip/hip_runtime.h>
#include <math.h>

typedef __attribute__((ext_vector_type(2))) float v2f;
typedef __attribute__((ext_vector_type(8))) float v8f;

#define DIM   23
#define DIM2  (DIM * DIM)       // 529
#define TRIU  (DIM * (DIM + 1) / 2)  // 276
#define BLOCK 128
#define STEP  1.0f

// One block per molecule. Phases:
//  1) stage X[n] (23x23) into LDS (NT loads, read-once data)
//  2) column sums-of-squares via V_WMMA_F32_16X16X4_F32 (A = X^2 transposed tile, B = ones)
//  3) stable argsort of (-norm + noise), 23 keys, rank-based
//  4) gather permuted upper triangle -> xr[276] in LDS
//  5) stream F soft-bin outputs with non-temporal stores (200MB write stream)
__global__ __launch_bounds__(BLOCK) void RandomSortCM_expand_kernel(
    const float* __restrict__ X,
    const float* __restrict__ noise,
    const int*   __restrict__ feat_idx,
    const float* __restrict__ thresholds,
    const float* __restrict__ mean,
    const float* __restrict__ stdp,
    float*       __restrict__ out,
    int F)
{
    __shared__ float xs[DIM2];
    __shared__ float normsq[32];
    __shared__ float key[DIM];
    __shared__ int   inds[DIM];
    __shared__ float xr[TRIU];

    const int n   = blockIdx.x;
    const int tid = threadIdx.x;
    const float* __restrict__ Xn = X + (size_t)n * DIM2;

    // ---- Phase 1: load X[n] to LDS (coalesced, non-temporal: each element read once) ----
    for (int t = tid; t < DIM2; t += BLOCK)
        xs[t] = __builtin_nontemporal_load(&Xn[t]);
    __syncthreads();

#if __has_builtin(__builtin_amdgcn_wmma_f32_16x16x4_f32)
    // ---- Phase 2: normsq[j] = sum_i xs[i*23+j]^2 via WMMA ----
    // D = A x B, B = all-ones (layout-independent), A[m,k] = xs[base+k, m]^2
    // => D[m,n] = partial column-m sum-of-squares, replicated over all n.
    // A 16x4 f32 layout: lanes 0-15 M=0..15 {VGPR0:K=0, VGPR1:K=1},
    //                    lanes 16-31 M=0..15 {VGPR0:K=2, VGPR1:K=3}.
    // All 4 waves execute the WMMA unpredicated (EXEC must be all 1s); results are
    // identical across waves, lanes 0 and 16 of the block write LDS.
    {
        const int lane  = tid & 31;
        const int m     = lane & 15;
        const int khalf = lane >> 4;          // 0 -> K{0,1}, 1 -> K{2,3}
        const int m2    = 16 + m;             // second M-tile: columns 16..22

        v2f bone; bone.x = 1.0f; bone.y = 1.0f;
        v8f acc0 = {0.f, 0.f, 0.f, 0.f, 0.f, 0.f, 0.f, 0.f};
        v8f acc1 = {0.f, 0.f, 0.f, 0.f, 0.f, 0.f, 0.f, 0.f};

        #pragma unroll
        for (int base = 0; base < 24; base += 4) {      // 6 K-chunks cover 23 rows (pad row 23)
            const int r0 = base + 2 * khalf;
            const int r1 = r0 + 1;
            float t0 = (r0 < DIM) ? xs[r0 * DIM + m] : 0.f;
            float t1 = (r1 < DIM) ? xs[r1 * DIM + m] : 0.f;
            v2f a0; a0.x = t0 * t0; a0.y = t1 * t1;
            acc0 = __builtin_amdgcn_wmma_f32_16x16x4_f32(
                false, a0, false, bone, (short)0, acc0, false, false);

            float u0 = (r0 < DIM && m2 < DIM) ? xs[r0 * DIM + m2] : 0.f;
            float u1 = (r1 < DIM && m2 < DIM) ? xs[r1 * DIM + m2] : 0.f;
            v2f a1; a1.x = u0 * u0; a1.y = u1 * u1;
            acc1 = __builtin_amdgcn_wmma_f32_16x16x4_f32(
                false, a1, false, bone, (short)0, acc1, false, false);
        }

        // C/D layout: VGPR r holds M=r (lanes 0-15) and M=8+r (lanes 16-31); value
        // independent of N here, so lane 0 / lane 16 carry everything we need.
        if (tid == 0) {
            #pragma unroll
            for (int r = 0; r < 8; ++r) normsq[r] = acc0[r];        // cols 0..7
            #pragma unroll
            for (int r = 0; r < 7; ++r) normsq[16 + r] = acc1[r];   // cols 16..22
        }
        if (tid == 16) {
            #pragma unroll
            for (int r = 0; r < 8; ++r) normsq[8 + r] = acc0[r];    // cols 8..15
        }
    }
#else
    // Fallback: plain VALU column reduction
    if (tid < DIM) {
        float s = 0.f;
        for (int i = 0; i < DIM; ++i) { float v = xs[i * DIM + tid]; s += v * v; }
        normsq[tid] = s;
    }
#endif
    __syncthreads();

    // ---- Phase 3: stable ascending argsort of key = -norm + noise ----
    if (tid < DIM)
        key[tid] = -sqrtf(normsq[tid]) + noise[(size_t)n * DIM + tid];
    __syncthreads();
    if (tid < DIM) {
        const float kj = key[tid];
        int r = 0;
        #pragma unroll
        for (int k = 0; k < DIM; ++k) {
            const float kk = key[k];
            r += (kk < kj) || (kk == kj && k < tid);   // stable tie-break
        }
        inds[r] = tid;
    }
    __syncthreads();

    // ---- Phase 4: permuted upper triangle (row-major triu order) ----
    for (int t = tid; t < TRIU; t += BLOCK) {
        int i = 0, rem = t;
        while (rem >= DIM - i) { rem -= DIM - i; ++i; }
        const int j = i + rem;
        xr[t] = xs[inds[i] * DIM + inds[j]];
    }
    __syncthreads();

    // ---- Phase 5: soft-bin expansion, streamed with NT stores ----
    const float inv_std = 1.0f / stdp[0];
    float* __restrict__ outn = out + (size_t)n * (size_t)F;
    for (int f = tid; f < F; f += BLOCK) {
        const int   fi = feat_idx[f];
        const float v  = tanhf((xr[fi] - thresholds[f]) * (1.0f / STEP));
        __builtin_nontemporal_store((v - mean[f]) * inv_std, &outn[f]);
    }
}

extern "C" void kernel_launch(void* const* d_in, const int* in_sizes, int n_in,
                              void* d_out, int out_size, void* d_ws, size_t ws_size,
                              hipStream_t stream) {
    (void)d_ws; (void)ws_size; (void)n_in; (void)out_size;
    const float* X          = (const float*)d_in[0];
    const float* noise      = (const float*)d_in[1];
    const int*   feat_idx   = (const int*)  d_in[2];
    const float* thresholds = (const float*)d_in[3];
    const float* mean       = (const float*)d_in[4];
    const float* stdp       = (const float*)d_in[5];
    float*       out        = (float*)d_out;

    const int N = in_sizes[1] / DIM;   // 16384
    const int F = in_sizes[2];         // total number of soft bins

    RandomSortCM_expand_kernel<<<N, BLOCK, 0, stream>>>(
        X, noise, feat_idx, thresholds, mean, stdp, out, F);
}